// CoordinationMemory_40183714021852
// MI455X (gfx1250) — compile-verified
//
#include <hip/hip_runtime.h>
#include <math.h>

typedef float f2 __attribute__((ext_vector_type(2)));
typedef float f4 __attribute__((ext_vector_type(4)));
typedef float f8 __attribute__((ext_vector_type(8)));

#define NROWS 4096
#define LV    64
#define HDIM  512
#define DDIM  256

// ---------------------------------------------------------------------------
// Kernel 1: streaming copy memory -> out (1 GiB traffic, the roofline floor).
// Non-temporal so the 512 MiB stream doesn't evict the 2.5 MiB of weights
// that kernel 2 wants resident in L2.
// ---------------------------------------------------------------------------
__global__ __launch_bounds__(256) void cm_copy(const f4* __restrict__ src,
                                               f4* __restrict__ dst,
                                               long n4) {
  long i = (long)blockIdx.x * blockDim.x + threadIdx.x;
  const long stride = (long)gridDim.x * blockDim.x;
  for (; i < n4; i += stride) {
    f4 v = __builtin_nontemporal_load(src + i);
    __builtin_nontemporal_store(v, dst + i);
  }
}

// ---------------------------------------------------------------------------
// Kernel 2: fused dual-GEMM + tanh + scatter using V_WMMA_F32_16X16X4_F32.
//   block = 256 threads = 8 wave32; grid = (N/16, H/128)
//   Each block: 16 rows of n, 128 columns of H (one 16-col tile per wave).
//   K = 1280 processed as 5 chunks of 256, each chunk staged 16x256 in LDS:
//     c=0 veh_repr, c=1 cust_repr, c=2 edge_emb, c=3,4 gathered cur_h halves.
// ---------------------------------------------------------------------------
__global__ __launch_bounds__(256) void cm_gemm(
    const float* __restrict__ memory,
    const int*   __restrict__ veh_idx,
    const float* __restrict__ veh_repr,
    const float* __restrict__ cust_repr,
    const float* __restrict__ edge_emb,
    const float* __restrict__ W_in,
    const float* __restrict__ b_in,
    const float* __restrict__ W_h,
    const float* __restrict__ b_h,
    float* __restrict__ out)
{
  __shared__ float As[16][264];   // 256 + 8 pad -> conflict-free column reads
  __shared__ int   idx_s[16];

  const int tid  = threadIdx.x;
  const int lane = tid & 31;
  const int wave = tid >> 5;
  const int n0 = blockIdx.x * 16;
  const int h0 = blockIdx.y * 128 + wave * 16;

  if (tid < 16) idx_s[tid] = veh_idx[n0 + tid];

  // A-matrix 16x4 f32 layout: lanes 0-15 -> K={0,1}, lanes 16-31 -> K={2,3}
  const int mrow = lane & 15;
  const int koff = (lane < 16) ? 0 : 2;

  f8 acc = {0.f, 0.f, 0.f, 0.f, 0.f, 0.f, 0.f, 0.f};

  for (int c = 0; c < 5; ++c) {
    __syncthreads();   // also covers idx_s before first use
    {
      // Stage 16 rows x 256 cols: thread t -> row t/16, four f4 per thread.
      const int r = tid >> 4;
      const int q = tid & 15;
      const int n = n0 + r;
      const float* src;
      if      (c == 0) src = veh_repr  + (long)n * DDIM;
      else if (c == 1) src = cust_repr + (long)n * DDIM;
      else if (c == 2) src = edge_emb  + (long)n * DDIM;
      else             src = memory + ((long)n * LV + idx_s[r]) * HDIM
                                     + (long)(c - 3) * DDIM;
#pragma unroll
      for (int j = 0; j < 4; ++j) {
        const int col = (q + j * 16) * 4;
        *(f4*)&As[r][col] = *(const f4*)(src + col);
      }
    }
    __syncthreads();

    // Row-major weight block for this K-chunk; B frag rows kb+koff, kb+koff+1.
    const float* Wp = (c < 3) ? (W_in + (long)(c * DDIM) * HDIM)
                              : (W_h  + (long)((c - 3) * DDIM) * HDIM);
    const float* bp = Wp + (long)koff * HDIM + h0 + mrow;

    for (int k4 = 0; k4 < DDIM / 4; ++k4) {
      const int kb = k4 * 4;
      f2 a = *(const f2*)&As[mrow][kb + koff];
      f2 b;
      b.x = bp[(long)kb * HDIM];
      b.y = bp[(long)(kb + 1) * HDIM];
      acc = __builtin_amdgcn_wmma_f32_16x16x4_f32(
          false, a, false, b, (short)0, acc, false, false);
    }
  }

  // Epilogue: bias + tanh, scatter into out[n, idx[n], h].
  // C/D layout: VGPR r holds M = r (lanes 0-15) or r+8 (lanes 16-31), N = lane&15.
  const int h = h0 + mrow;
  const float bias = b_in[h] + b_h[h];
  const int mbase = (lane < 16) ? 0 : 8;
#pragma unroll
  for (int r = 0; r < 8; ++r) {
    const int m = mbase + r;
    const int n = n0 + m;
    out[((long)n * LV + idx_s[m]) * HDIM + h] = tanhf(acc[r] + bias);
  }
}

extern "C" void kernel_launch(void* const* d_in, const int* in_sizes, int n_in,
                              void* d_out, int out_size, void* d_ws, size_t ws_size,
                              hipStream_t stream) {
  (void)in_sizes; (void)n_in; (void)out_size; (void)d_ws; (void)ws_size;

  const float* memory    = (const float*)d_in[0];
  const int*   veh_idx   = (const int*)  d_in[1];
  const float* veh_repr  = (const float*)d_in[2];
  const float* cust_repr = (const float*)d_in[3];
  const float* edge_emb  = (const float*)d_in[4];
  const float* W_in      = (const float*)d_in[5];
  const float* b_in      = (const float*)d_in[6];
  const float* W_h       = (const float*)d_in[7];
  const float* b_h       = (const float*)d_in[8];
  float* out = (float*)d_out;

  const long n4 = (long)NROWS * LV * HDIM / 4;   // 33.5M float4
  cm_copy<<<16384, 256, 0, stream>>>((const f4*)memory, (f4*)out, n4);

  dim3 grid(NROWS / 16, HDIM / 128);
  cm_gemm<<<grid, 256, 0, stream>>>(memory, veh_idx, veh_repr, cust_repr,
                                    edge_emb, W_in, b_in, W_h, b_h, out);
}